// VectorQuantizer_56315611185433
// MI455X (gfx1250) — compile-verified
//
#include <hip/hip_runtime.h>

// -------- problem constants (from reference) --------
#define NROWS        65536      // B*H*W = 16*64*64
#define DIM          16         // embedding dim
#define KCODES       1024       // num embeddings
#define ROWPAD       17         // LDS row pad: kills 16-stride bank conflicts
#define WAVES_PER_BLK 8
#define ROWS_PER_WAVE 16
#define ROWS_PER_BLK  (WAVES_PER_BLK * ROWS_PER_WAVE)   // 128
#define NBLK          (NROWS / ROWS_PER_BLK)            // 512
#define ZQ_ELEMS      (NROWS * DIM)                     // 1048576
#define LOSS_OFF      ZQ_ELEMS
#define IDX_OFF       (ZQ_ELEMS + 1)
#define LOSS_SCALE    (1.25f / (float)ZQ_ELEMS)         // (1 + 0.25) * mean

typedef float v2f __attribute__((ext_vector_type(2)));
typedef float v8f __attribute__((ext_vector_type(8)));

__global__ __launch_bounds__(256, 1)
void vq_wmma_kernel(const float* __restrict__ z,
                    const float* __restrict__ E,
                    float* __restrict__ out,
                    float* __restrict__ ws)
{
    __shared__ float sE[KCODES * ROWPAD];   // 69632 B, padded codebook
    __shared__ float sE2[KCODES];           // ||e_c||^2
    __shared__ float sred[256];             // loss reduction

    const int tid = threadIdx.x;

    // ---- stage codebook into LDS (coalesced float4 loads) ----
    for (int i = tid; i < (KCODES * DIM) / 4; i += 256) {
        float4 v  = ((const float4*)E)[i];
        int row   = (i * 4) >> 4;     // /DIM
        int col   = (i * 4) & 15;
        float* p  = &sE[row * ROWPAD + col];
        p[0] = v.x; p[1] = v.y; p[2] = v.z; p[3] = v.w;
    }
    __syncthreads();

    // ---- precompute ||e_c||^2 ----
    for (int c = tid; c < KCODES; c += 256) {
        float s = 0.f;
        #pragma unroll
        for (int k = 0; k < DIM; ++k) { float e = sE[c * ROWPAD + k]; s += e * e; }
        sE2[c] = s;
    }
    __syncthreads();

    const int wave   = tid >> 5;
    const int lane   = tid & 31;
    const int hi     = lane >> 4;     // which 16-lane half
    const int grp    = lane & 15;     // lane within half
    const int rowBase = blockIdx.x * ROWS_PER_BLK + wave * ROWS_PER_WAVE;

    // ---- A fragments: 16x16 z tile as 4 chained 16x4 f32 A-matrices ----
    // f32 A 16x4 layout: lanes 0-15 hold {K0,K1}, lanes 16-31 hold {K2,K3}
    v2f a[4];
    {
        const float* zrow = z + (size_t)(rowBase + grp) * DIM;
        #pragma unroll
        for (int i = 0; i < 4; ++i)
            a[i] = *(const v2f*)(zrow + 4 * i + 2 * hi);
    }

    float bestVal[8];
    int   bestIdx[8];
    #pragma unroll
    for (int r = 0; r < 8; ++r) { bestVal[r] = 3.4e38f; bestIdx[r] = 0; }

    // ---- sweep 64 code tiles of 16 codes each ----
    for (int t = 0; t < KCODES / 16; ++t) {
        const int   c   = t * 16 + grp;          // this lane's code column
        const float e2c = sE2[c];

        // f32 B 4x16 layout: VGPR0 = rows {K0 (lanes 0-15), K1 (lanes 16-31)},
        //                    VGPR1 = rows {K2, K3}; B[k][n] = E[n][k]
        v2f b[4];
        #pragma unroll
        for (int i = 0; i < 4; ++i) {
            b[i].x = sE[c * ROWPAD + 4 * i + hi];
            b[i].y = sE[c * ROWPAD + 4 * i + 2 + hi];
        }

        v8f acc = {};
        #pragma unroll
        for (int i = 0; i < 4; ++i) {
            // D = A(16x4,f32) x B(4x16,f32) + C ; exact fp32, K chained 4 at a time
            acc = __builtin_amdgcn_wmma_f32_16x16x4_f32(
                false, a[i], false, b[i], (short)0, acc, false, false);
        }

        // score = ||e||^2 - 2 z.e  (||z||^2 is row-constant, irrelevant to argmin)
        #pragma unroll
        for (int r = 0; r < 8; ++r) {
            float score = __builtin_fmaf(-2.0f, acc[r], e2c);
            if (score < bestVal[r]) { bestVal[r] = score; bestIdx[r] = c; }
        }
    }

    // ---- cross-lane argmin within each 16-lane half (wave32 butterflies) ----
    #pragma unroll
    for (int m = 1; m < 16; m <<= 1) {
        #pragma unroll
        for (int r = 0; r < 8; ++r) {
            float ov = __shfl_xor(bestVal[r], m, 32);
            int   oi = __shfl_xor(bestIdx[r], m, 32);
            if (ov < bestVal[r] || (ov == bestVal[r] && oi < bestIdx[r])) {
                bestVal[r] = ov; bestIdx[r] = oi;
            }
        }
    }

    // ---- emit z_q (gather from LDS), indices, and loss partial ----
    float lsum = 0.f;
    #pragma unroll
    for (int r = 0; r < 8; ++r) {
        const int row = rowBase + r + 8 * hi;
        const int idx = bestIdx[r];
        float q  = sE[idx * ROWPAD + grp];
        float zv = z[(size_t)row * DIM + grp];
        float d  = q - zv;
        lsum += d * d;
        out[(size_t)row * DIM + grp] = q;            // z_q_st forward value == z_q
        if (grp == 0) out[IDX_OFF + row] = (float)idx;
    }

    // ---- deterministic block reduction of loss partials ----
    sred[tid] = lsum;
    __syncthreads();
    for (int s = 128; s > 0; s >>= 1) {
        if (tid < s) sred[tid] += sred[tid + s];
        __syncthreads();
    }
    if (tid == 0) ws[blockIdx.x] = sred[0];
}

// Fixed-order final sum: replay-deterministic (no float atomics anywhere).
__global__ void vq_finalize_kernel(const float* __restrict__ ws,
                                   float* __restrict__ out)
{
    if (threadIdx.x == 0 && blockIdx.x == 0) {
        float s = 0.f;
        for (int i = 0; i < NBLK; ++i) s += ws[i];
        out[LOSS_OFF] = s * LOSS_SCALE;
    }
}

extern "C" void kernel_launch(void* const* d_in, const int* in_sizes, int n_in,
                              void* d_out, int out_size, void* d_ws, size_t ws_size,
                              hipStream_t stream)
{
    const float* z = (const float*)d_in[0];            // [65536, 16] fp32
    const float* E = (const float*)d_in[1];            // [1024, 16]  fp32
    float* out = (float*)d_out;                        // [zq | loss | idx]
    float* ws  = (float*)d_ws;                         // >= NBLK floats

    vq_wmma_kernel<<<NBLK, 256, 0, stream>>>(z, E, out, ws);
    vq_finalize_kernel<<<1, 32, 0, stream>>>(ws, out);
}